// Qfnn_2886218023530
// MI455X (gfx1250) — compile-verified
//
#include <hip/hip_runtime.h>
#include <math.h>

typedef __attribute__((ext_vector_type(2))) float v2f;
typedef __attribute__((ext_vector_type(8))) float v8f;

// workspace layout (floats)
#define WS_SUM   0     // [16]  sum of tnorm per channel
#define WS_SQ    16    // [16]  sum of tnorm^2 per channel
#define WS_URE   32    // [256] Re(U^T):  [k*16+n] = Re U[n][k]  (B-operand of WMMA1)
#define WS_UIM   288   // [256] Im(U^T)
#define WS_G     544   // [64]  G[o*16+b] = sum_j out_w[o,j]*sign_j(b)

#define TN_CAP 0.9999800001f   // (0.99999)^2, clip of s*s
#define ASTR   17              // padded LDS stride (conflict-free)

// ---------------------------------------------------------------------------
// per-row fuzzy t-norm (shared by stats pass and main pass)
// ---------------------------------------------------------------------------
__device__ __forceinline__ void compute_tnorm(const float* __restrict__ x4,
                                              const float* __restrict__ lin_w,
                                              const float* __restrict__ lin_b,
                                              const float* __restrict__ gn_g,
                                              const float* __restrict__ gn_b,
                                              const float* __restrict__ m,
                                              const float* __restrict__ theta,
                                              float tn[16]) {
    float xv[4];
#pragma unroll
    for (int k = 0; k < 4; ++k) xv[k] = x4[k];
    float h[4];
#pragma unroll
    for (int j = 0; j < 4; ++j) {
        float a = lin_b[j];
#pragma unroll
        for (int k = 0; k < 4; ++k) a += xv[k] * lin_w[j * 4 + k];
        h[j] = a;
    }
    float mu = 0.25f * (h[0] + h[1] + h[2] + h[3]);
    float var = 0.f;
#pragma unroll
    for (int j = 0; j < 4; ++j) { float d = h[j] - mu; var += d * d; }
    var *= 0.25f;
    float inv = rsqrtf(var + 1e-5f);
#pragma unroll
    for (int j = 0; j < 4; ++j) h[j] = (h[j] - mu) * inv * gn_g[j] + gn_b[j];

    // fuzzy memberships g[c][j], c in {0,1}; m, theta are (4,2) row-major
    float g[2][4];
#pragma unroll
    for (int j = 0; j < 4; ++j) {
#pragma unroll
        for (int c = 0; c < 2; ++c) {
            float d  = h[j] - m[j * 2 + c];
            float th = theta[j * 2 + c];
            float f  = __expf(-(d * d) / (2.f * th * th));
            g[c][j]  = fminf(f + 1e-16f, TN_CAP);
        }
    }
    // IDX[i][j] = (i >> (3-j)) & 1
#pragma unroll
    for (int i = 0; i < 16; ++i) {
        tn[i] = g[(i >> 3) & 1][0] * g[(i >> 2) & 1][1] *
                g[(i >> 1) & 1][2] * g[i & 1][3];
    }
}

// ---------------------------------------------------------------------------
// Kernel A: zero reduction buffers, fold circuit into U^T, fold signs+out_w
// into G.  1 block, 32 threads.
// ---------------------------------------------------------------------------
__global__ void qfnn_prep_kernel(const float* __restrict__ defuzz_w,
                                 const float* __restrict__ out_w,
                                 float* __restrict__ ws) {
    int t = threadIdx.x;
    if (t < 32) ws[t] = 0.f;   // zero sum / sumsq
    if (t >= 16) return;

    // thread t builds column t of U: circuit(e_t). qubit q <-> bit (3-q)
    float re[16], im[16];
#pragma unroll
    for (int n = 0; n < 16; ++n) { re[n] = 0.f; im[n] = 0.f; }
    re[t] = 1.f;

    for (int layer = 0; layer < 2; ++layer) {
        // CNOT chain: (0,1),(1,2),(2,3),(3,0)
        const int cq[4] = {0, 1, 2, 3};
        const int tq[4] = {1, 2, 3, 0};
        for (int gidx = 0; gidx < 4; ++gidx) {
            int mc = 1 << (3 - cq[gidx]);
            int mt = 1 << (3 - tq[gidx]);
            float nre[16], nim[16];
#pragma unroll
            for (int s = 0; s < 16; ++s) {
                int src = (s & mc) ? (s ^ mt) : s;
                nre[s] = re[src]; nim[s] = im[src];
            }
#pragma unroll
            for (int s = 0; s < 16; ++s) { re[s] = nre[s]; im[s] = nim[s]; }
        }
        // per-qubit RX, RZ, RX
        for (int q = 0; q < 4; ++q) {
            int mq = 1 << (3 - q);
            float th[3];
            th[0] = defuzz_w[layer * 12 + 3 * q + 0];
            th[1] = defuzz_w[layer * 12 + 3 * q + 1];
            th[2] = defuzz_w[layer * 12 + 3 * q + 2];
            for (int gi = 0; gi < 3; ++gi) {
                float c  = __cosf(0.5f * th[gi]);
                float sn = __sinf(0.5f * th[gi]);
#pragma unroll
                for (int s = 0; s < 16; ++s) {
                    if (s & mq) continue;          // s = |0> slot, p = |1> slot
                    int p = s | mq;
                    float ar = re[s], ai = im[s], br = re[p], bi = im[p];
                    if (gi == 1) {                 // RZ: a*=e^{-i t/2}, b*=e^{+i t/2}
                        re[s] = c * ar + sn * ai;  im[s] = c * ai - sn * ar;
                        re[p] = c * br - sn * bi;  im[p] = c * bi + sn * br;
                    } else {                       // RX: [[c,-i s],[-i s,c]]
                        re[s] = c * ar + sn * bi;  im[s] = c * ai - sn * br;
                        re[p] = c * br + sn * ai;  im[p] = c * bi - sn * ar;
                    }
                }
            }
        }
    }
    // store U^T:  B[k][n] = U[n][k] = (circuit(e_k))[n]
#pragma unroll
    for (int n = 0; n < 16; ++n) {
        ws[WS_URE + t * 16 + n] = re[n];
        ws[WS_UIM + t * 16 + n] = im[n];
    }
    // G[o][b] = sum_j out_w[o,j] * (1 - 2*bit_{3-j}(b));  thread t = basis b
#pragma unroll
    for (int o = 0; o < 4; ++o) {
        float acc = 0.f;
#pragma unroll
        for (int j = 0; j < 4; ++j) {
            float sgn = ((t >> (3 - j)) & 1) ? -1.f : 1.f;
            acc += out_w[o * 4 + j] * sgn;
        }
        ws[WS_G + o * 16 + t] = acc;
    }
}

// ---------------------------------------------------------------------------
// Kernel B: global per-channel sum / sumsq of tnorm (for the batch norm)
// ---------------------------------------------------------------------------
__global__ __launch_bounds__(256) void qfnn_stats_kernel(
        const float* __restrict__ x, const float* __restrict__ lin_w,
        const float* __restrict__ lin_b, const float* __restrict__ gn_g,
        const float* __restrict__ gn_b, const float* __restrict__ m,
        const float* __restrict__ theta, float* __restrict__ ws, int B) {
    __shared__ float s_acc[32];
    if (threadIdx.x < 32) s_acc[threadIdx.x] = 0.f;
    __syncthreads();

    float lsum[16], lsq[16];
#pragma unroll
    for (int i = 0; i < 16; ++i) { lsum[i] = 0.f; lsq[i] = 0.f; }

    for (int row = blockIdx.x * blockDim.x + threadIdx.x; row < B;
         row += gridDim.x * blockDim.x) {
        float tn[16];
        compute_tnorm(x + (size_t)row * 4, lin_w, lin_b, gn_g, gn_b, m, theta, tn);
#pragma unroll
        for (int i = 0; i < 16; ++i) { lsum[i] += tn[i]; lsq[i] += tn[i] * tn[i]; }
    }
#pragma unroll
    for (int i = 0; i < 16; ++i) {
        atomicAdd(&s_acc[i], lsum[i]);
        atomicAdd(&s_acc[16 + i], lsq[i]);
    }
    __syncthreads();
    if (threadIdx.x < 32) atomicAdd(&ws[threadIdx.x], s_acc[threadIdx.x]);
}

// ---------------------------------------------------------------------------
// Kernel C: fused batchnorm -> amp -> (WMMA) st=U*amp -> p=|st|^2
//           -> (WMMA) out = p @ G^T + out_b.
// Each wave32 handles 32 DISTINCT rows (= 2 WMMA tiles). All wave-invariant
// B-operands (U^T re/im, masked G) are hoisted into registers; A-operands
// come from padded LDS (conflict-free ds_load_2addr pairs).
// ---------------------------------------------------------------------------
__global__ __launch_bounds__(256) void qfnn_main_kernel(
        const float* __restrict__ x, const float* __restrict__ lin_w,
        const float* __restrict__ lin_b, const float* __restrict__ gn_g,
        const float* __restrict__ gn_b, const float* __restrict__ m,
        const float* __restrict__ theta, const float* __restrict__ bn_g,
        const float* __restrict__ bn_b, const float* __restrict__ out_b,
        const float* __restrict__ ws, float* __restrict__ out, int B) {
    __shared__ float lds_amp[8][32 * ASTR];  // 32 amp rows per wave, padded
    __shared__ float lds_p[8][16 * ASTR];    // one 16x16 p tile per wave, padded

    const int wave = threadIdx.x >> 5;
    const int lane = threadIdx.x & 31;
    const int hi   = lane >> 4;              // half-wave select
    const int l16  = lane & 15;
    const int base = blockIdx.x * 256 + wave * 32;   // first row of this wave

    int row  = base + lane;                  // one distinct row per lane
    int rowc = row < B ? row : (B - 1);

    // ---- preload wave-invariant WMMA B-operands (unconditional, in-bounds)
    const float* UreT = ws + WS_URE;
    const float* UimT = ws + WS_UIM;
    const float* G    = ws + WS_G;
    const float gmask = (l16 < 4) ? 1.f : 0.f;   // zero pad columns N>=4
    const int   gl    = (l16 & 3);               // clamped G row (always valid)
    v2f bre[4], bim[4], bg[4];
#pragma unroll
    for (int kk = 0; kk < 4; ++kk) {
        int kb = 4 * kk + 2 * hi;            // B: lane holds K=kb+{0,1}, N=l16
        bre[kk].x = UreT[kb * 16 + l16];  bre[kk].y = UreT[(kb + 1) * 16 + l16];
        bim[kk].x = UimT[kb * 16 + l16];  bim[kk].y = UimT[(kb + 1) * 16 + l16];
        bg[kk].x  = G[gl * 16 + kb]     * gmask;
        bg[kk].y  = G[gl * 16 + kb + 1] * gmask;
    }

    float tn[16];
    compute_tnorm(x + (size_t)rowc * 4, lin_w, lin_b, gn_g, gn_b, m, theta, tn);

    // batch norm + row normalization -> amplitudes, written straight to LDS
    const float invB = 1.f / (float)B;
    float* aw = lds_amp[wave];
    {
        float y[16];
        float ss = 0.f;
#pragma unroll
        for (int i = 0; i < 16; ++i) {
            float bm = ws[WS_SUM + i] * invB;
            float bv = ws[WS_SQ + i] * invB - bm * bm;
            float yi = (tn[i] - bm) * rsqrtf(bv + 1e-5f) * bn_g[i] + bn_b[i];
            y[i] = yi;
            ss += yi * yi;
        }
        float ninv = rsqrtf(ss);
#pragma unroll
        for (int i = 0; i < 16; ++i) aw[lane * ASTR + i] = y[i] * ninv;
    }
    __syncthreads();

    float* tp = lds_p[wave];

    for (int t = 0; t < 2; ++t) {            // two 16-row tiles per wave
        // ---- WMMA 1: st_re/st_im = amp(16x16) x U^T(16x16), K=16 via 4x K=4
        v8f dre = {0.f, 0.f, 0.f, 0.f, 0.f, 0.f, 0.f, 0.f};
        v8f dim = {0.f, 0.f, 0.f, 0.f, 0.f, 0.f, 0.f, 0.f};
#pragma unroll
        for (int kk = 0; kk < 4; ++kk) {
            int kb = 4 * kk + 2 * hi;        // A: lane holds M=l16, K=kb+{0,1}
            v2f a;
            a.x = aw[(t * 16 + l16) * ASTR + kb];
            a.y = aw[(t * 16 + l16) * ASTR + kb + 1];
            dre = __builtin_amdgcn_wmma_f32_16x16x4_f32(false, a, false, bre[kk],
                                                        (short)0, dre, false, false);
            dim = __builtin_amdgcn_wmma_f32_16x16x4_f32(false, a, false, bim[kk],
                                                        (short)0, dim, false, false);
        }

        // p = |st|^2 ; C/D layout: VGPR v -> M = v + 8*hi, N = l16
#pragma unroll
        for (int v = 0; v < 8; ++v) {
            float pv = dre[v] * dre[v] + dim[v] * dim[v];
            tp[(v + 8 * hi) * ASTR + l16] = pv;
        }
        __syncthreads();

        // ---- WMMA 2: out = p(16x16) x Gpad^T(16x16), only N<4 cols non-zero
        v8f dout = {0.f, 0.f, 0.f, 0.f, 0.f, 0.f, 0.f, 0.f};
#pragma unroll
        for (int kk = 0; kk < 4; ++kk) {
            int kb = 4 * kk + 2 * hi;
            v2f a2; a2.x = tp[l16 * ASTR + kb]; a2.y = tp[l16 * ASTR + kb + 1];
            dout = __builtin_amdgcn_wmma_f32_16x16x4_f32(false, a2, false, bg[kk],
                                                         (short)0, dout, false, false);
        }

        // store: lane with l16<4 owns output column l16 of rows v + 8*hi
        if (l16 < 4) {
            float ob = out_b[l16];
#pragma unroll
            for (int v = 0; v < 8; ++v) {
                int r = base + t * 16 + v + 8 * hi;
                if (r < B) out[(size_t)r * 4 + l16] = dout[v] + ob;
            }
        }
        __syncthreads();                     // p-tile reuse barrier (uniform)
    }
}

// ---------------------------------------------------------------------------
extern "C" void kernel_launch(void* const* d_in, const int* in_sizes, int n_in,
                              void* d_out, int out_size, void* d_ws, size_t ws_size,
                              hipStream_t stream) {
    const float* x        = (const float*)d_in[0];
    const float* lin_w    = (const float*)d_in[1];
    const float* lin_b    = (const float*)d_in[2];
    const float* gn_g     = (const float*)d_in[3];
    const float* gn_b     = (const float*)d_in[4];
    const float* m        = (const float*)d_in[5];
    const float* theta    = (const float*)d_in[6];
    // d_in[7] = q_w (unused by the reference)
    const float* bn_g     = (const float*)d_in[8];
    const float* bn_b     = (const float*)d_in[9];
    const float* defuzz_w = (const float*)d_in[10];
    const float* out_w    = (const float*)d_in[11];
    const float* out_b    = (const float*)d_in[12];
    float* ws  = (float*)d_ws;
    float* out = (float*)d_out;
    const int B = in_sizes[0] / 4;

    qfnn_prep_kernel<<<1, 32, 0, stream>>>(defuzz_w, out_w, ws);

    int sblocks = (B + 255) / 256;
    if (sblocks > 1024) sblocks = 1024;
    qfnn_stats_kernel<<<sblocks, 256, 0, stream>>>(x, lin_w, lin_b, gn_g, gn_b,
                                                   m, theta, ws, B);

    int mblocks = (B + 255) / 256;   // 8 waves/block, 32 rows per wave
    qfnn_main_kernel<<<mblocks, 256, 0, stream>>>(x, lin_w, lin_b, gn_g, gn_b,
                                                  m, theta, bn_g, bn_b, out_b,
                                                  ws, out, B);
}